// Model_58617713656334
// MI455X (gfx1250) — compile-verified
//
#include <hip/hip_runtime.h>
#include <math.h>

// ---------------------------------------------------------------------------
// FreTS/RWKV7 forecaster for MI455X (gfx1250, wave32).
// All dense GEMMs (DFT, projections, output proj, iDFT, FC head) go through
// V_WMMA_F32_16X16X4_F32 (fp32 WMMA, exact math). Each wave computes a 16x32
// output tile (A reused across two WMMAs per K-step) and prefetches upcoming
// A/B panels (global_prefetch_b8). The sequential RWKV-7 scan keeps its
// 169x169 per-batch state in LDS (CDNA5: 320KB/WGP).
// ---------------------------------------------------------------------------

#define BB 32     // batch
#define LL 336    // seq len
#define LP 352    // seq len padded to multiple of 32 (iDFT out / fc1 K)
#define CC 321    // channels (= RWKV time dim T)
#define DD 169    // rfft bins (= RWKV feature dim)
#define DPD 192   // DD padded to multiple of 32
#define HH 512    // hidden
#define PPRED 96  // pred len
#define RR (BB * CC)  // 10272 rows (multiple of 16)

typedef float v2f __attribute__((ext_vector_type(2)));
typedef float v8f __attribute__((ext_vector_type(8)));

// ---------------- fp32 WMMA GEMM: C[MxN] = A[MxK] * B[KxN] -----------------
// One wave per 16x32 output tile; K stepped by 4 via V_WMMA_F32_16X16X4_F32.
// A-layout: lane(m = lane&15, kh = lane>>4) holds A[m][k0+2kh], A[m][k0+2kh+1]
// B-layout: lane(n = lane&15, kh) holds B[k0+2kh][n], B[k0+2kh+1][n]
// C-layout: vgpr r on lane(n, mh) holds C[r + 8*mh][n]
// Requires: M % 16 == 0, N % 32 == 0, K % 16 == 0.
__global__ __launch_bounds__(32) void gemm_wmma_f32(
    const float* __restrict__ A, const float* __restrict__ B,
    float* __restrict__ C, int lda, int ldb, int ldc, int K) {
  const int lane = threadIdx.x;
  const int m = lane & 15;
  const int kh = lane >> 4;
  const long rowA = (long)blockIdx.y * 16 + m;
  const long col0 = (long)blockIdx.x * 32 + m;  // n index of left 16-tile
  const float* Ap = A + rowA * lda + 2 * kh;
  const float* Bp = B + (long)(2 * kh) * ldb + col0;
  v8f acc0 = {0.f, 0.f, 0.f, 0.f, 0.f, 0.f, 0.f, 0.f};
  v8f acc1 = {0.f, 0.f, 0.f, 0.f, 0.f, 0.f, 0.f, 0.f};
  for (int k0 = 0; k0 < K; k0 += 16) {
    // Speculative RT prefetch of the next K-panels (global_prefetch_b8).
    __builtin_prefetch((const void*)(Ap + 16), 0, 0);
    __builtin_prefetch((const void*)(Bp + 16 * (long)ldb), 0, 0);
#pragma unroll
    for (int kk = 0; kk < 4; ++kk) {
      v2f av, b0, b1;
      av.x = Ap[0];
      av.y = Ap[1];
      b0.x = Bp[0];
      b0.y = Bp[ldb];
      b1.x = Bp[16];
      b1.y = Bp[ldb + 16];
      acc0 = __builtin_amdgcn_wmma_f32_16x16x4_f32(false, av, false, b0,
                                                   (short)0, acc0, false,
                                                   false);
      acc1 = __builtin_amdgcn_wmma_f32_16x16x4_f32(false, av, false, b1,
                                                   (short)0, acc1, false,
                                                   false);
      Ap += 4;
      Bp += 4 * (long)ldb;
    }
  }
  float* Cp = C + ((long)blockIdx.y * 16 + 8 * kh) * ldc + col0;
#pragma unroll
  for (int r = 0; r < 8; ++r) {
    Cp[(long)r * ldc] = acc0[r];
    Cp[(long)r * ldc + 16] = acc1[r];
  }
}

// ------------------------------ helpers ------------------------------------
__device__ __forceinline__ float blockReduce(float v, float* buf, int tid,
                                             int nt) {
  buf[tid] = v;
  __syncthreads();
  for (int s = nt >> 1; s > 0; s >>= 1) {
    if (tid < s) buf[tid] += buf[tid + s];
    __syncthreads();
  }
  float r = buf[0];
  __syncthreads();
  return r;
}

// ---------------- RevIN normalize + transpose to [R][L] --------------------
__global__ __launch_bounds__(128) void revin_kernel(
    const float* __restrict__ x, const float* __restrict__ rw,
    const float* __restrict__ rb, float* __restrict__ zt,
    float* __restrict__ meanA, float* __restrict__ stdA) {
  __shared__ float buf[128];
  const int bc = blockIdx.x;
  const int b = bc / CC, c = bc % CC;
  const int tid = threadIdx.x;
  float s = 0.f, s2 = 0.f;
  for (int l = tid; l < LL; l += 128) {
    float v = x[((long)b * LL + l) * CC + c];
    s += v;
    s2 += v * v;
  }
  float S1 = blockReduce(s, buf, tid, 128);
  float S2 = blockReduce(s2, buf, tid, 128);
  float mean = S1 / (float)LL;
  float var = S2 / (float)LL - mean * mean;
  float sd = sqrtf(var + 1e-5f);
  if (tid == 0) {
    meanA[bc] = mean;
    stdA[bc] = sd;
  }
  const float g = rw[c], be = rb[c];
  for (int l = tid; l < LL; l += 128)
    zt[(long)bc * LL + l] =
        (x[((long)b * LL + l) * CC + c] - mean) / sd * g + be;
}

// -------- forward DFT table [L][2*DPD]: cols 0..DPD-1 cos, DPD.. -sin ------
__global__ void fwdtab_kernel(float* __restrict__ T) {
  int i = blockIdx.x * 256 + threadIdx.x;
  if (i >= LL * 2 * DPD) return;
  int l = i / (2 * DPD), col = i % (2 * DPD);
  int d = (col < DPD) ? col : col - DPD;
  float v = 0.f;
  if (d < DD) {
    float th = 6.283185307179586f * (float)((d * l) % LL) / (float)LL;
    float sn, cs;
    __sincosf(th, &sn, &cs);
    const float isl = rsqrtf((float)LL);  // ortho norm
    v = (col < DPD) ? cs * isl : -sn * isl;
  }
  T[i] = v;
}

// ---- inverse DFT table [2*DPD][LP]: rows 0..DPD-1 for Re, DPD.. for Im ----
__global__ void idfttab_kernel(float* __restrict__ T) {
  int i = blockIdx.x * 256 + threadIdx.x;
  if (i >= 2 * DPD * LP) return;
  int r = i / LP, l = i % LP;
  int half = r / DPD, d = r % DPD;
  float v = 0.f;
  if (d < DD && l < LL) {
    const float isl = rsqrtf((float)LL);
    bool edge = (d == 0) || (d == DD - 1);  // DC / Nyquist
    float th = 6.283185307179586f * (float)((d * l) % LL) / (float)LL;
    float sn, cs;
    __sincosf(th, &sn, &cs);
    if (half == 0)
      v = (edge ? 1.f : 2.f) * cs * isl;
    else
      v = edge ? 0.f : -2.f * sn * isl;  // Im of DC/Nyquist discarded (c2r)
  }
  T[i] = v;
}

// ---------------- zero-pad a DDxDD weight into DPDxDPD ---------------------
__global__ void padw_kernel(const float* __restrict__ src,
                            float* __restrict__ dst) {
  int i = blockIdx.x * 256 + threadIdx.x;
  if (i >= DPD * DPD) return;
  int r = i / DPD, c = i % DPD;
  dst[i] = (r < DD && c < DD) ? src[r * DD + c] : 0.f;
}

// ---------------- zero-pad fc_w1 [LL x HH] into [LP x HH] ------------------
__global__ void padfc1_kernel(const float* __restrict__ src,
                              float* __restrict__ dst) {
  int i = blockIdx.x * 256 + threadIdx.x;
  if (i >= LP * HH) return;
  int r = i / HH, c = i % HH;
  dst[i] = (r < LL) ? src[r * HH + c] : 0.f;
}

// ---------------- LayerNorm over DD of one half of xf ----------------------
__global__ __launch_bounds__(256) void ln_kernel(const float* __restrict__ xf,
                                                 const float* __restrict__ g,
                                                 const float* __restrict__ b2,
                                                 float* __restrict__ h,
                                                 int half) {
  __shared__ float buf[256];
  const int row = blockIdx.x, tid = threadIdx.x;
  const float* xr = xf + (long)row * 2 * DPD + half * DPD;
  float s = 0.f;
  for (int j = tid; j < DD; j += 256) s += xr[j];
  float mean = blockReduce(s, buf, tid, 256) / (float)DD;
  float d2 = 0.f;
  for (int j = tid; j < DD; j += 256) {
    float t = xr[j] - mean;
    d2 += t * t;
  }
  float var = blockReduce(d2, buf, tid, 256) / (float)DD;
  float inv = rsqrtf(var + 1e-5f);
  float* hr = h + (long)row * DPD;
  for (int j = tid; j < DPD; j += 256)
    hr[j] = (j < DD) ? (xr[j] - mean) * inv * g[j] + b2[j] : 0.f;
}

// ------- elementwise: w=exp(-softplus), a=sigmoid, kk=k/(||k||+1e-6) -------
__global__ __launch_bounds__(256) void postproj_kernel(
    float* __restrict__ pw, float* __restrict__ pa,
    const float* __restrict__ pk, float* __restrict__ pkk) {
  __shared__ float buf[256];
  const int row = blockIdx.x, tid = threadIdx.x;
  const float* kr = pk + (long)row * DPD;
  float s = 0.f;
  for (int j = tid; j < DD; j += 256) {
    float kv = kr[j];
    s += kv * kv;
  }
  float inv = 1.f / (sqrtf(blockReduce(s, buf, tid, 256)) + 1e-6f);
  float* wr_ = pw + (long)row * DPD;
  float* ar = pa + (long)row * DPD;
  float* kkr = pkk + (long)row * DPD;
  for (int j = tid; j < DPD; j += 256) {
    if (j < DD) {
      float wl = wr_[j];
      float sp = (wl > 20.f) ? wl : log1pf(expf(wl));  // softplus
      wr_[j] = expf(-sp);
      ar[j] = 1.f / (1.f + expf(-ar[j]));
      kkr[j] = kr[j] * inv;
    } else {
      wr_[j] = 0.f;
      ar[j] = 0.f;
      kkr[j] = 0.f;
    }
  }
}

// -------- RWKV-7 generalized delta-rule scan; S[169x169] in LDS ------------
// One workgroup per batch. Thread i owns row i of S (exclusive), so only the
// per-step vector staging needs barriers.
__global__ __launch_bounds__(256) void rwkv7_scan_kernel(
    const float* __restrict__ pr, const float* __restrict__ pk,
    const float* __restrict__ pv, const float* __restrict__ pw,
    const float* __restrict__ pa, const float* __restrict__ pkk,
    float* __restrict__ yraw) {
  extern __shared__ float sm[];
  const int SROW = 170;  // avoid bank conflicts
  float* S = sm;
  float* vR = sm + DD * SROW;
  float* vK = vR + DPD;
  float* vV = vK + DPD;
  float* vW = vV + DPD;
  float* vA = vW + DPD;
  float* vKK = vA + DPD;
  const int b = blockIdx.x, tid = threadIdx.x;
  for (int i = tid; i < DD * SROW; i += 256) S[i] = 0.f;
  __syncthreads();
  for (int t = 0; t < CC; ++t) {
    const long row = (long)b * CC + t;
    const float* rp = pr + row * DPD;
    const float* kp = pk + row * DPD;
    const float* vp = pv + row * DPD;
    const float* wp = pw + row * DPD;
    const float* ap = pa + row * DPD;
    const float* kkp = pkk + row * DPD;
    for (int j = tid; j < DD; j += 256) {
      vR[j] = rp[j];
      vK[j] = kp[j];
      vV[j] = vp[j];
      vW[j] = wp[j];
      vA[j] = ap[j];
      vKK[j] = kkp[j];
    }
    __syncthreads();
    if (tid < DD) {
      float* Si = S + tid * SROW;
      float sa = 0.f;
      for (int j = 0; j < DD; ++j) sa += Si[j] * vKK[j];
      const float vi = vV[tid];
      float yi = 0.f;
      for (int j = 0; j < DD; ++j) {
        float sv = Si[j] * vW[j] - sa * vKK[j] * vA[j] + vi * vK[j];
        Si[j] = sv;
        yi += sv * vR[j];
      }
      yraw[row * DPD + tid] = yi;
    } else if (tid < DPD) {
      yraw[row * DPD + tid] = 0.f;  // keep pads zeroed
    }
    __syncthreads();
  }
}

// ---------------- residual add into packed yb half -------------------------
__global__ void addres_kernel(const float* __restrict__ xf,
                              const float* __restrict__ tmp,
                              float* __restrict__ yb, int half) {
  long i = (long)blockIdx.x * 256 + threadIdx.x;
  if (i >= (long)RR * DPD) return;
  long row = i / DPD;
  int col = (int)(i % DPD);
  long idx = row * 2 * DPD + half * DPD + col;
  yb[idx] = xf[idx] + tmp[i];
}

// ---------------- bias + LeakyReLU(0.01) on hfc ----------------------------
__global__ void biasrelu_kernel(float* __restrict__ hfc,
                                const float* __restrict__ b1) {
  long i = (long)blockIdx.x * 256 + threadIdx.x;
  if (i >= (long)RR * HH) return;
  float v = hfc[i] + b1[(int)(i % HH)];
  hfc[i] = (v >= 0.f) ? v : 0.01f * v;
}

// ------- final: bias2, RevIN denorm, transpose to [B][PRED][C] -------------
__global__ void final_kernel(const float* __restrict__ outb,
                             const float* __restrict__ b2,
                             const float* __restrict__ rw,
                             const float* __restrict__ rb,
                             const float* __restrict__ stdA,
                             const float* __restrict__ meanA,
                             float* __restrict__ out) {
  long i = (long)blockIdx.x * 256 + threadIdx.x;
  if (i >= (long)BB * PPRED * CC) return;
  int c = (int)(i % CC);
  int p = (int)((i / CC) % PPRED);
  int b = (int)(i / ((long)CC * PPRED));
  float v = outb[((long)b * CC + c) * PPRED + p] + b2[p];
  int bc = b * CC + c;
  out[i] = (v - rb[c]) / (rw[c] + 1e-10f) * stdA[bc] + meanA[bc];
}

// ---------------------------------------------------------------------------
static inline void gemm(const float* A, const float* B, float* C, int M, int N,
                        int K, int lda, int ldb, int ldc, hipStream_t st) {
  dim3 grid(N / 32, M / 16);
  gemm_wmma_f32<<<grid, 32, 0, st>>>(A, B, C, lda, ldb, ldc, K);
}

extern "C" void kernel_launch(void* const* d_in, const int* in_sizes, int n_in,
                              void* d_out, int out_size, void* d_ws,
                              size_t ws_size, hipStream_t stream) {
  (void)in_sizes;
  (void)n_in;
  (void)out_size;
  (void)ws_size;
  const float* x = (const float*)d_in[0];
  const float* revin_w = (const float*)d_in[4];
  const float* revin_b = (const float*)d_in[5];
  const float* ln_g[2] = {(const float*)d_in[6], (const float*)d_in[14]};
  const float* ln_b[2] = {(const float*)d_in[7], (const float*)d_in[15]};
  const float* wmats[2][6];  // wr wk wv ww wa wo
  for (int p = 0; p < 2; ++p)
    for (int i = 0; i < 6; ++i) wmats[p][i] = (const float*)d_in[8 + p * 8 + i];
  const float* fc_w1 = (const float*)d_in[22];
  const float* fc_b1 = (const float*)d_in[23];
  const float* fc_w2 = (const float*)d_in[24];
  const float* fc_b2 = (const float*)d_in[25];
  float* out = (float*)d_out;

  // ---- workspace carve-up (floats) ----
  float* ws = (float*)d_ws;
  size_t o = 0;
  float* MEAN = ws + o;  o += RR;
  float* STDA = ws + o;  o += RR;
  float* ZT   = ws + o;  o += (size_t)RR * LP;        // zt (ld=LL); later YTIME (ld=LP)
  float* FWDT = ws + o;  o += (size_t)LL * 2 * DPD;
  float* IDFT = ws + o;  o += (size_t)2 * DPD * LP;
  float* FC1P = ws + o;  o += (size_t)LP * HH;
  float* XF   = ws + o;  o += (size_t)RR * 2 * DPD;   // [real | imag] per row
  float* HBUF = ws + o;  o += (size_t)RR * DPD;       // LN out / wo-GEMM tmp
  float* PRJ  = ws + o;  o += (size_t)RR * DPD * 6;   // later aliased: HFC
  float* YRAW = ws + o;  o += (size_t)RR * DPD;       // later aliased: OUTB
  float* YB   = ws + o;  o += (size_t)RR * 2 * DPD;
  float* WPAD = ws + o;  o += (size_t)12 * DPD * DPD;
  float* PR = PRJ + (size_t)0 * RR * DPD;
  float* PK = PRJ + (size_t)1 * RR * DPD;
  float* PV = PRJ + (size_t)2 * RR * DPD;
  float* PW = PRJ + (size_t)3 * RR * DPD;
  float* PA = PRJ + (size_t)4 * RR * DPD;
  float* PKK = PRJ + (size_t)5 * RR * DPD;
  float* YTIME = ZT;   // zt dead after forward DFT
  float* HFC = PRJ;    // projections dead after both scans
  float* OUTB = YRAW;  // yraw dead after wo GEMMs

  // 1) RevIN + transpose
  revin_kernel<<<RR, 128, 0, stream>>>(x, revin_w, revin_b, ZT, MEAN, STDA);

  // 2) DFT tables + padded weights
  fwdtab_kernel<<<(LL * 2 * DPD + 255) / 256, 256, 0, stream>>>(FWDT);
  idfttab_kernel<<<(2 * DPD * LP + 255) / 256, 256, 0, stream>>>(IDFT);
  padfc1_kernel<<<(LP * HH + 255) / 256, 256, 0, stream>>>(fc_w1, FC1P);
  for (int p = 0; p < 2; ++p)
    for (int i = 0; i < 6; ++i)
      padw_kernel<<<(DPD * DPD + 255) / 256, 256, 0, stream>>>(
          wmats[p][i], WPAD + (size_t)(p * 6 + i) * DPD * DPD);

  // 3) forward rDFT: XF[R x 2*DPD] = ZT[R x LL] * FWDT[LL x 2*DPD]
  gemm(ZT, FWDT, XF, RR, 2 * DPD, LL, LL, 2 * DPD, 2 * DPD, stream);

  // 4) RWKV-7 on real (part 0) and imag (part 1) halves
  const size_t smem = (size_t)(DD * 170 + 6 * DPD) * sizeof(float);
  for (int part = 0; part < 2; ++part) {
    float* WP = WPAD + (size_t)part * 6 * DPD * DPD;
    ln_kernel<<<RR, 256, 0, stream>>>(XF, ln_g[part], ln_b[part], HBUF, part);
    float* projs[5] = {PR, PK, PV, PW, PA};
    for (int i = 0; i < 5; ++i)  // h @ {wr,wk,wv,ww,wa}
      gemm(HBUF, WP + (size_t)i * DPD * DPD, projs[i], RR, DPD, DPD, DPD, DPD,
           DPD, stream);
    postproj_kernel<<<RR, 256, 0, stream>>>(PW, PA, PK, PKK);
    rwkv7_scan_kernel<<<BB, 256, smem, stream>>>(PR, PK, PV, PW, PA, PKK,
                                                 YRAW);
    // y @ wo -> HBUF, then residual into packed YB half
    gemm(YRAW, WP + (size_t)5 * DPD * DPD, HBUF, RR, DPD, DPD, DPD, DPD, DPD,
         stream);
    addres_kernel<<<(int)(((long)RR * DPD + 255) / 256), 256, 0, stream>>>(
        XF, HBUF, YB, part);
  }

  // 5) inverse rDFT: YTIME[R x LP] = YB[R x 2*DPD] * IDFT[2*DPD x LP]
  //    (cols >= LL are zero; overwrites the dead ZT region)
  gemm(YB, IDFT, YTIME, RR, LP, 2 * DPD, 2 * DPD, LP, LP, stream);

  // 6) FC head (fc_w1 zero-padded to LP rows so padded time cols are inert)
  gemm(YTIME, FC1P, HFC, RR, HH, LP, LP, HH, HH, stream);
  biasrelu_kernel<<<(int)(((long)RR * HH + 255) / 256), 256, 0, stream>>>(
      HFC, fc_b1);
  gemm(HFC, fc_w2, OUTB, RR, PPRED, HH, HH, PPRED, PPRED, stream);

  // 7) bias2 + RevIN denorm + transpose to [B][PRED][C]
  final_kernel<<<(int)(((long)BB * PPRED * CC + 255) / 256), 256, 0, stream>>>(
      OUTB, fc_b2, revin_w, revin_b, STDA, MEAN, out);
}